// CSIH_6339371728953
// MI455X (gfx1250) — compile-verified
//
#include <hip/hip_runtime.h>
#include <math.h>

// ---------------- problem constants (match reference) ----------------
#define N_USERS 100000
#define N_ITEMS 50000
#define NTOT    150000      // N_USERS + N_ITEMS
#define DIM     128
#define KP      8           // intent prototypes
#define E_G     1500000
#define NL      65536
#define EL      262144
#define BB      1024

typedef float v2f __attribute__((ext_vector_type(2)));
typedef float v8f __attribute__((ext_vector_type(8)));

static inline int cdiv(int a, int b) { return (a + b - 1) / b; }

__device__ __forceinline__ void atomAddF(float* p, float v) {
    unsafeAtomicAdd(p, v);   // -> global_atomic_add_f32 on gfx1250
}

// =====================================================================
// Generic fp32 WMMA GEMM:  C[M,Nc] = act(A[M,K] @ W[K,Nc] + bias)
// One wave32 per 16x16 output tile; K-loop of V_WMMA_F32_16X16X4_F32.
// A 16x4 layout: lanes0-15 row M=l (K=k0,k0+1), lanes16-31 row M=l-16 (K=k0+2,k0+3)
// B 4x16 layout: mirrored; C/D: VGPR r -> (M=r + 8*half, N=lane&15).
// =====================================================================
__global__ void gemm_wmma_f32(const float* __restrict__ A, int lda,
                              const float* __restrict__ W, int ldw,
                              const float* __restrict__ bias,
                              float* __restrict__ C, int ldc,
                              int M, int K, int Nc, int act)
{
    const int mbase = blockIdx.x * 16;
    const int nbase = blockIdx.y * 16;
    const int lane  = threadIdx.x;      // 0..31
    const int half  = lane >> 4;        // 0/1
    const int lr    = lane & 15;

    int rowA = mbase + lr; if (rowA >= M)  rowA = M - 1;   // clamp, store-guarded
    int colB = nbase + lr; if (colB >= Nc) colB = Nc - 1;

    const float* ap = A + (size_t)rowA * lda;
    v8f acc = {0.f, 0.f, 0.f, 0.f, 0.f, 0.f, 0.f, 0.f};

    for (int k0 = 0; k0 < K; k0 += 4) {
        const int kA = k0 + 2 * half;
        v2f a, b;
        a.x = ap[kA];
        a.y = ap[kA + 1];
        b.x = W[(size_t)kA * ldw + colB];
        b.y = W[(size_t)(kA + 1) * ldw + colB];
        acc = __builtin_amdgcn_wmma_f32_16x16x4_f32(false, a, false, b,
                                                    (short)0, acc, false, false);
    }

    const int col = nbase + lr;
    if (col < Nc) {
        const float bv = bias ? bias[col] : 0.f;
#pragma unroll
        for (int r = 0; r < 8; ++r) {
            const int row = mbase + r + 8 * half;
            if (row < M) {
                float v = acc[r] + bv;
                if (act == 1)      v = tanhf(v);
                else if (act == 2) v = fmaxf(v, 0.f);
                C[(size_t)row * ldc + col] = v;
            }
        }
    }
}

// Same GEMM but A rows gathered from two embedding tables (all_emb concat).
__global__ void gemm_wmma_2tab(const float* __restrict__ T0,   // user_emb
                               const float* __restrict__ T1,   // item_emb
                               const float* __restrict__ W, int ldw,
                               const float* __restrict__ bias,
                               float* __restrict__ C, int ldc,
                               int M, int K, int Nc)
{
    const int mbase = blockIdx.x * 16;
    const int nbase = blockIdx.y * 16;
    const int lane  = threadIdx.x;
    const int half  = lane >> 4;
    const int lr    = lane & 15;

    int rowA = mbase + lr; if (rowA >= M)  rowA = M - 1;
    int colB = nbase + lr; if (colB >= Nc) colB = Nc - 1;

    const float* ap = (rowA < N_USERS) ? (T0 + (size_t)rowA * DIM)
                                       : (T1 + (size_t)(rowA - N_USERS) * DIM);
    v8f acc = {0.f, 0.f, 0.f, 0.f, 0.f, 0.f, 0.f, 0.f};

    for (int k0 = 0; k0 < K; k0 += 4) {
        const int kA = k0 + 2 * half;
        v2f a, b;
        a.x = ap[kA];
        a.y = ap[kA + 1];
        b.x = W[(size_t)kA * ldw + colB];
        b.y = W[(size_t)(kA + 1) * ldw + colB];
        acc = __builtin_amdgcn_wmma_f32_16x16x4_f32(false, a, false, b,
                                                    (short)0, acc, false, false);
    }

    const int col = nbase + lr;
    if (col < Nc) {
        const float bv = bias ? bias[col] : 0.f;
#pragma unroll
        for (int r = 0; r < 8; ++r) {
            const int row = mbase + r + 8 * half;
            if (row < M) C[(size_t)row * ldc + col] = acc[r] + bv;
        }
    }
}

// ---------------- elementwise / scatter kernels ----------------
__global__ void zero_f32(float* __restrict__ p, size_t n) {
    size_t i = (size_t)blockIdx.x * blockDim.x + threadIdx.x;
    if (i < n) p[i] = 0.f;
}

// sign-weighted message scatter: aggr[dst] += sign_emb[attr] * xw_all[src]
__global__ void edge_scatter_sign(const int* __restrict__ gei,
                                  const int* __restrict__ gea,
                                  const float* __restrict__ sign_emb,
                                  const float* __restrict__ xw,
                                  float* __restrict__ aggr)
{
    size_t t = (size_t)blockIdx.x * blockDim.x + threadIdx.x;
    int e    = (int)(t >> 5);
    int lane = (int)(t & 31);
    if (e >= E_G) return;
    const int src = gei[e];
    const int dst = gei[E_G + e];
    const float s = sign_emb[gea[e]];
    const float4 v = *((const float4*)(xw + (size_t)src * DIM) + lane);
    float* dp = aggr + (size_t)dst * DIM + lane * 4;
    atomAddF(dp + 0, s * v.x);
    atomAddF(dp + 1, s * v.y);
    atomAddF(dp + 2, s * v.z);
    atomAddF(dp + 3, s * v.w);
}

// global_repr = all_emb + relu(aggr)   (in place over aggr)
__global__ void add_relu_emb(const float* __restrict__ ue,
                             const float* __restrict__ ie,
                             float* __restrict__ aggr)
{
    size_t i = (size_t)blockIdx.x * blockDim.x + threadIdx.x;
    if (i >= (size_t)NTOT * DIM) return;
    const int row = (int)(i >> 7);
    const float e = (row < N_USERS) ? ue[i] : ie[i - (size_t)N_USERS * DIM];
    aggr[i] = e + fmaxf(aggr[i], 0.f);
}

// gu[b,:] = repr[root_u[b]], gi[b,:] = repr[root_i[b]+N_USERS]
__global__ void gather_roots(const float* __restrict__ repr,
                             const int* __restrict__ ru,
                             const int* __restrict__ ri,
                             float* __restrict__ gu, float* __restrict__ gi)
{
    size_t i = (size_t)blockIdx.x * blockDim.x + threadIdx.x;
    if (i >= (size_t)BB * DIM) return;
    const int b = (int)(i >> 7), c = (int)(i & 127);
    gu[i] = repr[(size_t)ru[b] * DIM + c];
    gi[i] = repr[((size_t)ri[b] + N_USERS) * DIM + c];
}

__global__ void softmax8(const float* __restrict__ logits, float* __restrict__ probs) {
    int b = blockIdx.x * blockDim.x + threadIdx.x;
    if (b >= BB) return;
    float l[KP], m = -1e30f;
#pragma unroll
    for (int k = 0; k < KP; ++k) { l[k] = logits[b * KP + k]; m = fmaxf(m, l[k]); }
    float s = 0.f;
#pragma unroll
    for (int k = 0; k < KP; ++k) { l[k] = expf(l[k] - m); s += l[k]; }
    const float inv = 1.f / s;
#pragma unroll
    for (int k = 0; k < KP; ++k) probs[b * KP + k] = l[k] * inv;
}

// loc_in[i, 0:128]=repr[x_idx[i]], loc_in[i,128:256]=dist_emb[dist_label[i]]
__global__ void build_local_in(const float* __restrict__ repr,
                               const float* __restrict__ dist_emb,
                               const int* __restrict__ x_idx,
                               const int* __restrict__ dlab,
                               float* __restrict__ loc_in)
{
    size_t i = (size_t)blockIdx.x * blockDim.x + threadIdx.x;
    if (i >= (size_t)NL * 256) return;
    const int row = (int)(i >> 8), j = (int)(i & 255);
    loc_in[i] = (j < DIM) ? repr[(size_t)x_idx[row] * DIM + j]
                          : dist_emb[(size_t)dlab[row] * DIM + (j - DIM)];
}

__global__ void deg_init(float* __restrict__ deg) {       // self-loop
    int i = blockIdx.x * blockDim.x + threadIdx.x;
    if (i < NL) deg[i] = 1.f;
}
__global__ void deg_edges(const int* __restrict__ lei, float* __restrict__ deg) {
    int e = blockIdx.x * blockDim.x + threadIdx.x;
    if (e < EL) atomAddF(&deg[lei[EL + e]], 1.f);
}
__global__ void deg_to_dinv(float* __restrict__ deg) {    // deg>=1 always
    int i = blockIdx.x * blockDim.x + threadIdx.x;
    if (i < NL) deg[i] = rsqrtf(deg[i]);
}

// acc[dst] += dinv[src]*dinv[dst] * xw[src]
__global__ void gcn_scatter(const int* __restrict__ lei,
                            const float* __restrict__ dinv,
                            const float* __restrict__ xw,
                            float* __restrict__ acc)
{
    size_t t = (size_t)blockIdx.x * blockDim.x + threadIdx.x;
    int e    = (int)(t >> 5);
    int lane = (int)(t & 31);
    if (e >= EL) return;
    const int src = lei[e];
    const int dst = lei[EL + e];
    const float nrm = dinv[src] * dinv[dst];
    const float4 v = *((const float4*)(xw + (size_t)src * DIM) + lane);
    float* dp = acc + (size_t)dst * DIM + lane * 4;
    atomAddF(dp + 0, nrm * v.x);
    atomAddF(dp + 1, nrm * v.y);
    atomAddF(dp + 2, nrm * v.z);
    atomAddF(dp + 3, nrm * v.w);
}

// local_out = acc + dinv^2 * xw (self loop) + gcn_b   (in place over acc)
__global__ void gcn_finalize(const float* __restrict__ dinv,
                             const float* __restrict__ xw,
                             const float* __restrict__ gcn_b,
                             float* __restrict__ acc)
{
    size_t i = (size_t)blockIdx.x * blockDim.x + threadIdx.x;
    if (i >= (size_t)NL * DIM) return;
    const int row = (int)(i >> 7), c = (int)(i & 127);
    const float d = dinv[row];
    acc[i] = acc[i] + d * d * xw[i] + gcn_b[c];
}

__global__ void pool_count(const int* __restrict__ batch, float* __restrict__ counts) {
    int i = blockIdx.x * blockDim.x + threadIdx.x;
    if (i < NL) atomAddF(&counts[batch[i]], 1.f);
}
__global__ void pool_sum(const int* __restrict__ batch,
                         const float* __restrict__ lo, float* __restrict__ hsum)
{
    size_t i = (size_t)blockIdx.x * blockDim.x + threadIdx.x;
    if (i >= (size_t)NL * DIM) return;
    const int row = (int)(i >> 7), c = (int)(i & 127);
    atomAddF(&hsum[(size_t)batch[row] * DIM + c], lo[i]);
}
__global__ void pool_mean(const float* __restrict__ hsum,
                          const float* __restrict__ counts, float* __restrict__ hpre)
{
    size_t i = (size_t)blockIdx.x * blockDim.x + threadIdx.x;
    if (i >= (size_t)BB * DIM) return;
    const int b = (int)(i >> 7);
    hpre[i] = hsum[i] / fmaxf(counts[b], 1.f);
}

// final_vec = [h_sub | gu | gi | ru*rv]  (B x 512)
__global__ void build_final_vec(const float* __restrict__ h_sub,
                                const float* __restrict__ gu,
                                const float* __restrict__ gi,
                                const float* __restrict__ ru,
                                const float* __restrict__ rv,
                                float* __restrict__ fvec)
{
    size_t i = (size_t)blockIdx.x * blockDim.x + threadIdx.x;
    if (i >= (size_t)BB * 4 * DIM) return;
    const int b = (int)(i >> 9), j = (int)(i & 511);
    const size_t o = (size_t)b * DIM;
    float v;
    if (j < 128)       v = h_sub[o + j];
    else if (j < 256)  v = gu[o + (j - 128)];
    else if (j < 384)  v = gi[o + (j - 256)];
    else               v = ru[o + (j - 384)] * rv[o + (j - 384)];
    fvec[i] = v;
}

__global__ void final_pred(const float* __restrict__ hidden,
                           const float* __restrict__ w2,
                           const float* __restrict__ b2,
                           float* __restrict__ pred)
{
    int b = blockIdx.x * blockDim.x + threadIdx.x;
    if (b >= BB) return;
    float s = b2[0];
#pragma unroll
    for (int k = 0; k < 64; ++k) s += hidden[b * 64 + k] * w2[k];
    pred[b] = 1.f / (1.f + expf(-s));
}

// =====================================================================
extern "C" void kernel_launch(void* const* d_in, const int* in_sizes, int n_in,
                              void* d_out, int out_size, void* d_ws, size_t ws_size,
                              hipStream_t stream)
{
    // ---- inputs (setup_inputs order) ----
    const int*   gei      = (const int*)  d_in[0];
    const int*   gea      = (const int*)  d_in[1];
    const int*   root_u   = (const int*)  d_in[2];
    const int*   root_i   = (const int*)  d_in[3];
    const int*   x_idx    = (const int*)  d_in[4];
    const int*   dlab     = (const int*)  d_in[5];
    const int*   lei      = (const int*)  d_in[6];
    const int*   batch    = (const int*)  d_in[7];
    const float* user_emb = (const float*)d_in[8];
    const float* item_emb = (const float*)d_in[9];
    const float* dist_emb = (const float*)d_in[10];
    const float* c_u      = (const float*)d_in[11];
    const float* c_v      = (const float*)d_in[12];
    const float* gnn_W    = (const float*)d_in[13];
    const float* gnn_b    = (const float*)d_in[14];
    const float* sign_emb = (const float*)d_in[15];
    const float* iw1      = (const float*)d_in[16];
    const float* ib1      = (const float*)d_in[17];
    const float* iw2      = (const float*)d_in[18];
    const float* ib2      = (const float*)d_in[19];
    const float* lpw      = (const float*)d_in[20];
    const float* lpb      = (const float*)d_in[21];
    const float* gcn_w    = (const float*)d_in[22];
    const float* gcn_b    = (const float*)d_in[23];
    const float* pool_w   = (const float*)d_in[24];
    const float* pool_b   = (const float*)d_in[25];
    const float* fw1      = (const float*)d_in[26];
    const float* fb1      = (const float*)d_in[27];
    const float* fw2      = (const float*)d_in[28];
    const float* fb2      = (const float*)d_in[29];

    float* ws = (float*)d_ws;
    float* out = (float*)d_out;

    // ---- d_out layout: pred | gu | h_sub | pu | pv ----
    float* pred  = out;
    float* gu    = out + 1024;
    float* h_sub = out + 1024 + 131072;
    float* pu    = out + 1024 + 2 * 131072;
    float* pv    = out + 1024 + 2 * 131072 + 8192;

    // ---- workspace layout (floats); ~160 MB total ----
    const size_t SZ_BIG = (size_t)NTOT * DIM;          // 19,200,000
    float* regA   = ws;                                 // xw_all -> loc_in -> gcn acc
    float* regB   = ws + SZ_BIG;                        // aggr -> global_repr -> local_h/xw_local
    float* smallD = ws + 2 * SZ_BIG;
    float* gi      = smallD;                  // 131072
    float* t1      = smallD + 131072;         // 131072
    float* logits  = smallD + 262144;         // 8192
    float* ru      = smallD + 270336;         // 131072
    float* rv      = smallD + 401408;         // 131072
    float* deg     = smallD + 532480;         // 65536 (becomes dinv)
    float* hsum    = smallD + 598016;         // 131072
    float* counts  = smallD + 729088;         // 1024
    float* hpre    = smallD + 730112;         // 131072
    float* fvec    = smallD + 861184;         // 524288
    float* hidden  = smallD + 1385472;        // 65536

    float* xw_all   = regA;
    float* aggr     = regB;                   // then global_repr in place
    float* loc_in   = regA;                   // NL x 256 (after xw_all dead)
    float* local_h  = regB;                   // NL x 128 (after repr dead)
    float* xw_local = regB + (size_t)NL * DIM;
    float* gcn_acc  = regA;                   // NL x 128 (after loc_in dead)

    const dim3 b32(32, 1, 1);
    const int TB = 256;

    // 1) zero aggr
    zero_f32<<<cdiv((int)SZ_BIG, TB), TB, 0, stream>>>(aggr, SZ_BIG);

    // 2) xw_all = all_emb @ gnn_W + gnn_b      [150000,128]@[128,128]
    gemm_wmma_2tab<<<dim3(NTOT / 16, DIM / 16), b32, 0, stream>>>(
        user_emb, item_emb, gnn_W, DIM, gnn_b, xw_all, DIM, NTOT, DIM, DIM);

    // 3) edge scatter (sign-weighted messages)
    edge_scatter_sign<<<cdiv(E_G * 32, TB), TB, 0, stream>>>(gei, gea, sign_emb, xw_all, aggr);

    // 4) global_repr = all_emb + relu(aggr)  (in place)
    add_relu_emb<<<cdiv((int)SZ_BIG, TB), TB, 0, stream>>>(user_emb, item_emb, aggr);
    float* repr = aggr;

    // 5) gather roots (gu straight into d_out)
    gather_roots<<<cdiv(BB * DIM, TB), TB, 0, stream>>>(repr, root_u, root_i, gu, gi);

    // 6) intent(u):  t1=tanh(gu@iw1+ib1); logits=t1@iw2+ib2; pu=softmax; ru=pu@c_u
    gemm_wmma_f32<<<dim3(BB / 16, DIM / 16), b32, 0, stream>>>(gu, DIM, iw1, DIM, ib1, t1, DIM, BB, DIM, DIM, 1);
    gemm_wmma_f32<<<dim3(BB / 16, 1), b32, 0, stream>>>(t1, DIM, iw2, KP, ib2, logits, KP, BB, DIM, KP, 0);
    softmax8<<<cdiv(BB, TB), TB, 0, stream>>>(logits, pu);
    gemm_wmma_f32<<<dim3(BB / 16, DIM / 16), b32, 0, stream>>>(pu, KP, c_u, DIM, nullptr, ru, DIM, BB, KP, DIM, 0);

    // 7) intent(v)
    gemm_wmma_f32<<<dim3(BB / 16, DIM / 16), b32, 0, stream>>>(gi, DIM, iw1, DIM, ib1, t1, DIM, BB, DIM, DIM, 1);
    gemm_wmma_f32<<<dim3(BB / 16, 1), b32, 0, stream>>>(t1, DIM, iw2, KP, ib2, logits, KP, BB, DIM, KP, 0);
    softmax8<<<cdiv(BB, TB), TB, 0, stream>>>(logits, pv);
    gemm_wmma_f32<<<dim3(BB / 16, DIM / 16), b32, 0, stream>>>(pv, KP, c_v, DIM, nullptr, rv, DIM, BB, KP, DIM, 0);

    // 8) local features: concat(repr[x_idx], dist_emb[dist_label]) -> proj
    build_local_in<<<cdiv(NL * 256, TB), TB, 0, stream>>>(repr, dist_emb, x_idx, dlab, loc_in);
    gemm_wmma_f32<<<dim3(NL / 16, DIM / 16), b32, 0, stream>>>(loc_in, 2 * DIM, lpw, DIM, lpb, local_h, DIM, NL, 2 * DIM, DIM, 0);

    // 9) GCN: xw_local = local_h @ gcn_w (bias added after aggregation)
    gemm_wmma_f32<<<dim3(NL / 16, DIM / 16), b32, 0, stream>>>(local_h, DIM, gcn_w, DIM, nullptr, xw_local, DIM, NL, DIM, DIM, 0);

    //    degrees (with self loops) -> dinv
    deg_init<<<cdiv(NL, TB), TB, 0, stream>>>(deg);
    deg_edges<<<cdiv(EL, TB), TB, 0, stream>>>(lei, deg);
    deg_to_dinv<<<cdiv(NL, TB), TB, 0, stream>>>(deg);

    //    normalized scatter + self loop + bias
    zero_f32<<<cdiv(NL * DIM, TB), TB, 0, stream>>>(gcn_acc, (size_t)NL * DIM);
    gcn_scatter<<<cdiv(EL * 32, TB), TB, 0, stream>>>(lei, deg, xw_local, gcn_acc);
    gcn_finalize<<<cdiv(NL * DIM, TB), TB, 0, stream>>>(deg, xw_local, gcn_b, gcn_acc);

    // 10) mean pool -> tanh(pool)
    zero_f32<<<cdiv(131072 + 1024, TB), TB, 0, stream>>>(hsum, 131072 + 1024); // hsum + counts contiguous
    pool_count<<<cdiv(NL, TB), TB, 0, stream>>>(batch, counts);
    pool_sum<<<cdiv(NL * DIM, TB), TB, 0, stream>>>(batch, gcn_acc, hsum);
    pool_mean<<<cdiv(BB * DIM, TB), TB, 0, stream>>>(hsum, counts, hpre);
    gemm_wmma_f32<<<dim3(BB / 16, DIM / 16), b32, 0, stream>>>(hpre, DIM, pool_w, DIM, pool_b, h_sub, DIM, BB, DIM, DIM, 1);

    // 11) head: fvec=[h_sub|gu|gi|ru*rv]; hidden=relu(fvec@fw1+fb1); pred=sigmoid(...)
    build_final_vec<<<cdiv(BB * 4 * DIM, TB), TB, 0, stream>>>(h_sub, gu, gi, ru, rv, fvec);
    gemm_wmma_f32<<<dim3(BB / 16, 64 / 16), b32, 0, stream>>>(fvec, 4 * DIM, fw1, 64, fb1, hidden, 64, BB, 4 * DIM, 64, 2);
    final_pred<<<cdiv(BB, TB), TB, 0, stream>>>(hidden, fw2, fb2, pred);

    (void)in_sizes; (void)n_in; (void)out_size; (void)ws_size;
}